// GeneratorNet_2972117369003
// MI455X (gfx1250) — compile-verified
//
#include <hip/hip_runtime.h>
#include <hip/hip_bf16.h>

// ---------------------------------------------------------------------------
// CDNA5 (gfx1250) generator net, fp32 WMMA (V_WMMA_F32_16X16X4_F32).
// All GEMM-shaped work (convT taps, adjacency edge/proj) runs on WMMA with
// 32x32 tiles per wave (4 accumulators, A/B fragment reuse -> 2 loads/WMMA).
// ---------------------------------------------------------------------------

typedef __attribute__((ext_vector_type(2))) float v2f;
typedef __attribute__((ext_vector_type(8))) float v8f;

// D = A(16x4) x B(4x16) + C, fp32 tensor op (8-arg VOP3P form)
#define WMMA4F32(a, b, c) \
  __builtin_amdgcn_wmma_f32_16x16x4_f32(false, (a), false, (b), (short)0, (c), false, false)

// Fragment layouts (ISA 7.12.2, wave32):
//  A 16x4 : lane = 16*kh + m (m=0..15, kh=K-half), reg r -> K = 2*kh + r
//  B 4x16 : lane = 16*kh + n,                      reg r -> K = 2*kh + r
//  C 16x16: reg r, lane -> (M = r + 8*(lane>=16), N = lane&15)

// ---------------------------------------------------------------------------
// ConvT tap GEMM:  U[k,b] = W_k^T (cout x cin) @ X_b (cin x Lin)
//   A[m,kk] = wt[(kk*cout + m)*4 + k]   (strided read of (cin,cout,4) tensor)
// grid: (Lin/32, cout/32, 64=4taps*16batch), block 32 (one wave, 32x32 tile)
// ---------------------------------------------------------------------------
__global__ __launch_bounds__(32)
void convt_tap_gemm(const float* __restrict__ wt, const float* __restrict__ x,
                    float* __restrict__ U, int cin, int cout, int Lin)
{
  const int lane = threadIdx.x & 31;
  const int mr = lane & 15, kh = lane >> 4;
  const int k = blockIdx.z >> 4, b = blockIdx.z & 15;
  const int m0 = blockIdx.y * 32, n0 = blockIdx.x * 32;
  const int col0 = n0 + mr, col1 = col0 + 16;

  const float* X = x + (size_t)b * cin * Lin;
  const float* Alo = wt + (size_t)(m0 + mr) * 4 + k;       // + kk*(cout*4) per K
  const float* Ahi = wt + (size_t)(m0 + 16 + mr) * 4 + k;
  const size_t a_cs = (size_t)cout * 4;

  v8f acc00 = {}, acc01 = {}, acc10 = {}, acc11 = {};
  for (int kk = 0; kk < cin; kk += 4) {
    const int ka = kk + 2 * kh;
    v2f a0, a1, b0, b1;
    a0.x = Alo[(size_t)ka * a_cs];       a0.y = Alo[(size_t)(ka + 1) * a_cs];
    a1.x = Ahi[(size_t)ka * a_cs];       a1.y = Ahi[(size_t)(ka + 1) * a_cs];
    b0.x = X[(size_t)ka * Lin + col0];   b0.y = X[(size_t)(ka + 1) * Lin + col0];
    b1.x = X[(size_t)ka * Lin + col1];   b1.y = X[(size_t)(ka + 1) * Lin + col1];
    acc00 = WMMA4F32(a0, b0, acc00);
    acc01 = WMMA4F32(a0, b1, acc01);
    acc10 = WMMA4F32(a1, b0, acc10);
    acc11 = WMMA4F32(a1, b1, acc11);
  }
  float* D = U + ((size_t)(k * 16 + b) * cout + m0) * Lin;
  for (int r = 0; r < 8; ++r) {
    const int m = r + 8 * kh;
    D[(size_t)m * Lin + col0]        = acc00[r];
    D[(size_t)m * Lin + col1]        = acc01[r];
    D[(size_t)(m + 16) * Lin + col0] = acc10[r];
    D[(size_t)(m + 16) * Lin + col1] = acc11[r];
  }
}

// ---------------------------------------------------------------------------
// Adjacency edge GEMM (per batch):
//   edge = W0 @ X[:,src] + W1 @ X[:,dst]   (C x E)
//   A0[m,kk]=wadj[(m*C+kk)*2], A1 = +1; gather fused into B-fragment loads.
// grid: (E/32, C/32), block 32
// ---------------------------------------------------------------------------
__global__ __launch_bounds__(32)
void adj_edge_gemm(const float* __restrict__ wadj, const float* __restrict__ x,
                   const int* __restrict__ src, const int* __restrict__ dst,
                   float* __restrict__ edge, int C, int Ll, int E)
{
  const int lane = threadIdx.x & 31;
  const int mr = lane & 15, kh = lane >> 4;
  const int m0 = blockIdx.y * 32, n0 = blockIdx.x * 32;
  const int e0 = n0 + mr, e1 = e0 + 16;
  const int cs0 = src[e0], cs1 = src[e1];
  const int cd0 = dst[e0], cd1 = dst[e1];

  const float* Alo = wadj + (size_t)(m0 + mr) * C * 2;
  const float* Ahi = wadj + (size_t)(m0 + 16 + mr) * C * 2;

  v8f acc00 = {}, acc01 = {}, acc10 = {}, acc11 = {};
  for (int kk = 0; kk < C; kk += 4) {
    const int ka = kk + 2 * kh;
    v2f a00, a01, a10, a11, bs0, bs1, bd0, bd1;
    a00.x = Alo[(size_t)ka * 2];     a00.y = Alo[(size_t)(ka + 1) * 2];      // W0 rows m0..
    a01.x = Alo[(size_t)ka * 2 + 1]; a01.y = Alo[(size_t)(ka + 1) * 2 + 1];  // W1 rows m0..
    a10.x = Ahi[(size_t)ka * 2];     a10.y = Ahi[(size_t)(ka + 1) * 2];      // W0 rows m0+16..
    a11.x = Ahi[(size_t)ka * 2 + 1]; a11.y = Ahi[(size_t)(ka + 1) * 2 + 1];  // W1 rows m0+16..
    bs0.x = x[(size_t)ka * Ll + cs0]; bs0.y = x[(size_t)(ka + 1) * Ll + cs0];
    bs1.x = x[(size_t)ka * Ll + cs1]; bs1.y = x[(size_t)(ka + 1) * Ll + cs1];
    bd0.x = x[(size_t)ka * Ll + cd0]; bd0.y = x[(size_t)(ka + 1) * Ll + cd0];
    bd1.x = x[(size_t)ka * Ll + cd1]; bd1.y = x[(size_t)(ka + 1) * Ll + cd1];
    acc00 = WMMA4F32(a00, bs0, acc00);
    acc00 = WMMA4F32(a01, bd0, acc00);
    acc01 = WMMA4F32(a00, bs1, acc01);
    acc01 = WMMA4F32(a01, bd1, acc01);
    acc10 = WMMA4F32(a10, bs0, acc10);
    acc10 = WMMA4F32(a11, bd0, acc10);
    acc11 = WMMA4F32(a10, bs1, acc11);
    acc11 = WMMA4F32(a11, bd1, acc11);
  }
  float* D = edge + (size_t)m0 * E;
  for (int r = 0; r < 8; ++r) {
    const int m = r + 8 * kh;
    D[(size_t)m * E + e0]        = acc00[r];
    D[(size_t)m * E + e1]        = acc01[r];
    D[(size_t)(m + 16) * E + e0] = acc10[r];
    D[(size_t)(m + 16) * E + e1] = acc11[r];
  }
}

// ---------------------------------------------------------------------------
// Adjacency projection + scatter (per batch):
//   P = W_which^T (C x C) @ edge (C x E);  out[:, idx[e]] += P[:, e]  (atomic)
//   A[m,kk] = wadj[(kk*C + m)*2 + which]
// grid: (E/32, C/32), block 32
// ---------------------------------------------------------------------------
__global__ __launch_bounds__(32)
void adj_proj_scatter(const float* __restrict__ wadj, int which,
                      const float* __restrict__ edge, const int* __restrict__ idx,
                      float* __restrict__ out, int C, int Ll, int E)
{
  const int lane = threadIdx.x & 31;
  const int mr = lane & 15, kh = lane >> 4;
  const int m0 = blockIdx.y * 32, n0 = blockIdx.x * 32;
  const int col0 = n0 + mr, col1 = col0 + 16;

  v8f acc00 = {}, acc01 = {}, acc10 = {}, acc11 = {};
  for (int kk = 0; kk < C; kk += 4) {
    const int ka = kk + 2 * kh;
    v2f a0, a1, b0, b1;
    a0.x = wadj[((size_t)ka * C + (m0 + mr)) * 2 + which];
    a0.y = wadj[((size_t)(ka + 1) * C + (m0 + mr)) * 2 + which];
    a1.x = wadj[((size_t)ka * C + (m0 + 16 + mr)) * 2 + which];
    a1.y = wadj[((size_t)(ka + 1) * C + (m0 + 16 + mr)) * 2 + which];
    b0.x = edge[(size_t)ka * E + col0];
    b0.y = edge[(size_t)(ka + 1) * E + col0];
    b1.x = edge[(size_t)ka * E + col1];
    b1.y = edge[(size_t)(ka + 1) * E + col1];
    acc00 = WMMA4F32(a0, b0, acc00);
    acc01 = WMMA4F32(a0, b1, acc01);
    acc10 = WMMA4F32(a1, b0, acc10);
    acc11 = WMMA4F32(a1, b1, acc11);
  }
  const int p0 = idx[col0], p1 = idx[col1];
  for (int r = 0; r < 8; ++r) {
    const int m = m0 + r + 8 * kh;
    atomicAdd(&out[(size_t)m * Ll + p0],        acc00[r]);
    atomicAdd(&out[(size_t)m * Ll + p1],        acc01[r]);
    atomicAdd(&out[(size_t)(m + 16) * Ll + p0], acc10[r]);
    atomicAdd(&out[(size_t)(m + 16) * Ll + p1], acc11[r]);
  }
}

// ----------------------------- elementwise ---------------------------------
__global__ void zero_f32(float* __restrict__ p, size_t n) {
  size_t i = (size_t)blockIdx.x * blockDim.x + threadIdx.x;
  if (i < n) p[i] = 0.0f;
}

// x[b,o,n] = relu(sum_f z[b, n*50+f] * w[o*50+f]);  out layout (16,2048,32)
__global__ void linear_relu(const float* __restrict__ z, const float* __restrict__ w,
                            float* __restrict__ x)
{
  size_t id = (size_t)blockIdx.x * blockDim.x + threadIdx.x;
  if (id >= (size_t)16 * 2048 * 32) return;
  const int n = id & 31;
  const int o = (id >> 5) & 2047;
  const int b = (int)(id >> 16);
  const float* zp = z + (size_t)b * 1600 + n * 50;
  const float* wp = w + (size_t)o * 50;
  float s = 0.f;
  for (int f = 0; f < 50; ++f) s += zp[f] * wp[f];
  x[id] = fmaxf(s, 0.f);
}

// y[b,co,t] from taps: t=2s+k-1.  U: (4,16,cout,Lin)
__global__ void convt_combine(const float* __restrict__ U, float* __restrict__ y,
                              int cout, int Lin)
{
  const int Lout = 2 * Lin;
  size_t id = (size_t)blockIdx.x * blockDim.x + threadIdx.x;
  if (id >= (size_t)16 * cout * Lout) return;
  const int t = (int)(id % Lout);
  const int co = (int)((id / Lout) % cout);
  const int b = (int)(id / ((size_t)Lout * cout));
  const size_t base = ((size_t)b * cout + co) * Lin;
  const size_t tap = (size_t)16 * cout * Lin;
  float v = 0.f;
  if (t & 1) {
    const int s0 = (t + 1) >> 1, s2 = (t - 1) >> 1;
    if (s0 < Lin) v += U[0 * tap + base + s0];
    v += U[2 * tap + base + s2];
  } else {
    const int s1 = t >> 1, s3 = s1 - 1;
    v += U[1 * tap + base + s1];
    if (s3 >= 0) v += U[3 * tap + base + s3];
  }
  y[id] = v;
}

__global__ void count_deg(const int* __restrict__ src, const int* __restrict__ dst,
                          float* __restrict__ deg, int E)
{
  int e = blockIdx.x * blockDim.x + threadIdx.x;
  if (e >= E) return;
  atomicAdd(&deg[src[e]], 1.0f);
  atomicAdd(&deg[dst[e]], 1.0f);
}

__global__ void div_deg(float* __restrict__ x, const float* __restrict__ deg,
                        int Ll, size_t total)
{
  size_t i = (size_t)blockIdx.x * blockDim.x + threadIdx.x;
  if (i >= total) return;
  x[i] /= fmaxf(deg[i % Ll], 1.0f);
}

// per (b,c) row of length L: relu((x-mu)*rsqrt(var+eps)); grid = 16*C blocks
__global__ void instnorm_relu(float* __restrict__ x, int L)
{
  float* row = x + (size_t)blockIdx.x * L;
  __shared__ float s1[256], s2[256];
  const int tid = threadIdx.x;
  float sum = 0.f, sq = 0.f;
  for (int t = tid; t < L; t += 256) { float v = row[t]; sum += v; sq += v * v; }
  s1[tid] = sum; s2[tid] = sq;
  __syncthreads();
  for (int sft = 128; sft > 0; sft >>= 1) {
    if (tid < sft) { s1[tid] += s1[tid + sft]; s2[tid] += s2[tid + sft]; }
    __syncthreads();
  }
  const float mu = s1[0] / (float)L;
  const float var = s2[0] / (float)L - mu * mu;
  const float inv = rsqrtf(var + 1e-5f);
  for (int t = tid; t < L; t += 256) row[t] = fmaxf(0.f, (row[t] - mu) * inv);
}

// softmax over 32 channels at each (b,t); x: (16,32,2048)
__global__ void softmax_ch(const float* __restrict__ x, float* __restrict__ out)
{
  int id = blockIdx.x * blockDim.x + threadIdx.x;
  if (id >= 16 * 2048) return;
  const int b = id >> 11, t = id & 2047;
  const float* p = x + (size_t)b * 32 * 2048 + t;
  float v[32];
  float mx = -INFINITY;
  for (int c = 0; c < 32; ++c) { v[c] = p[(size_t)c * 2048]; mx = fmaxf(mx, v[c]); }
  float s = 0.f;
  for (int c = 0; c < 32; ++c) { v[c] = __expf(v[c] - mx); s += v[c]; }
  float* o = out + (size_t)b * 32 * 2048 + t;
  const float inv = 1.0f / s;
  for (int c = 0; c < 32; ++c) o[(size_t)c * 2048] = v[c] * inv;
}

// ---------------------------------------------------------------------------
extern "C" void kernel_launch(void* const* d_in, const int* in_sizes, int n_in,
                              void* d_out, int out_size, void* d_ws, size_t ws_size,
                              hipStream_t stream)
{
  (void)in_sizes; (void)n_in; (void)out_size; (void)ws_size;
  const float* z     = (const float*)d_in[0];
  const int* srcs[4] = {(const int*)d_in[1], (const int*)d_in[3],
                        (const int*)d_in[5], (const int*)d_in[7]};
  const int* dsts[4] = {(const int*)d_in[2], (const int*)d_in[4],
                        (const int*)d_in[6], (const int*)d_in[8]};
  const float* w_lin = (const float*)d_in[9];
  const float* wts[6] = {(const float*)d_in[10], (const float*)d_in[11],
                         (const float*)d_in[12], (const float*)d_in[13],
                         (const float*)d_in[14], (const float*)d_in[15]};
  const float* wadjs[4] = {(const float*)d_in[16], (const float*)d_in[17],
                           (const float*)d_in[18], (const float*)d_in[19]};
  float* out = (float*)d_out;

  // workspace (floats): xA(1M) xB(1M) U(2M) edge(2M) deg(2K)  ~ 25 MB
  float* xA   = (float*)d_ws;
  float* xB   = xA + (1u << 20);
  float* U    = xB + (1u << 20);
  float* edge = U  + (1u << 21);
  float* deg  = edge + (1u << 21);

  linear_relu<<<(16 * 2048 * 32 + 255) / 256, 256, 0, stream>>>(z, w_lin, xA);

  float *cur = xA, *oth = xB;
  int cin = 2048, Lin = 32;
  for (int i = 1; i <= 6; ++i) {
    const int cout = cin / 2;
    const int Lout = 2 * Lin;

    // 4 tap GEMMs, all taps x all batches in one grid (32x32 tiles)
    dim3 g1(Lin / 32, cout / 32, 64);
    convt_tap_gemm<<<g1, 32, 0, stream>>>(wts[i - 1], cur, U, cin, cout, Lin);

    const size_t ytot = (size_t)16 * cout * Lout;
    convt_combine<<<(unsigned)((ytot + 255) / 256), 256, 0, stream>>>(U, oth, cout, Lin);
    { float* t = cur; cur = oth; oth = t; }  // cur = convT output

    if (i >= 3) {
      const int l = 6 - i;
      const int C = cout, Ll = Lout, E = Ll * 32;
      const size_t tot = (size_t)16 * C * Ll;
      zero_f32<<<(unsigned)((tot + 255) / 256), 256, 0, stream>>>(oth, tot);
      zero_f32<<<(Ll + 255) / 256, 256, 0, stream>>>(deg, (size_t)Ll);
      count_deg<<<(E + 255) / 256, 256, 0, stream>>>(srcs[l], dsts[l], deg, E);
      dim3 g2(E / 32, C / 32, 1);
      for (int b = 0; b < 16; ++b) {
        const float* xb = cur + (size_t)b * C * Ll;
        float* ob = oth + (size_t)b * C * Ll;
        adj_edge_gemm<<<g2, 32, 0, stream>>>(wadjs[l], xb, srcs[l], dsts[l], edge, C, Ll, E);
        adj_proj_scatter<<<g2, 32, 0, stream>>>(wadjs[l], 0, edge, srcs[l], ob, C, Ll, E);
        adj_proj_scatter<<<g2, 32, 0, stream>>>(wadjs[l], 1, edge, dsts[l], ob, C, Ll, E);
      }
      div_deg<<<(unsigned)((tot + 255) / 256), 256, 0, stream>>>(oth, deg, Ll, tot);
      { float* t = cur; cur = oth; oth = t; }  // cur = adjacency output
    }

    if (i <= 5) instnorm_relu<<<16 * cout, 256, 0, stream>>>(cur, Lout);

    cin = cout; Lin = Lout;
  }

  softmax_ch<<<(16 * 2048 + 255) / 256, 256, 0, stream>>>(cur, out);
}